// FeedForwardQuantum_65481071396395
// MI455X (gfx1250) — compile-verified
//
#include <hip/hip_runtime.h>
#include <math.h>

typedef float v2f __attribute__((ext_vector_type(2)));
typedef float v8f __attribute__((ext_vector_type(8)));

#define EDIM  1024
#define FDIM  12
#define FPAD  16
#define W2S   1040          // padded row stride (floats) for W2^T in LDS (bank-conflict free)
#define WAVES 8             // waves per block
#define BLOCKS 128          // 128 blocks * 8 waves = 1024 waves = 1024 16-token tiles

// One WMMA f32 16x16x4 step: D = A(16x4) * B(4x16) + C
#define G_STEP(acc, avec, bx, by)                                              \
  {                                                                            \
    v2f _b; _b.x = (bx); _b.y = (by);                                          \
    (acc) = __builtin_amdgcn_wmma_f32_16x16x4_f32(                             \
        false, (avec), false, _b, (short)0, (acc), false, false);              \
  }

__global__ __launch_bounds__(256, 1)
void ffq_fused_wmma(const float* __restrict__ x,
                    const float* __restrict__ W1,
                    const float* __restrict__ b1,
                    const float* __restrict__ theta,
                    const float* __restrict__ W2,
                    const float* __restrict__ b2,
                    float* __restrict__ out)
{
  __shared__ float w1t[EDIM * FPAD];   // 64 KB : w1t[e*16 + f], f 12..15 zeroed
  __shared__ float w2t[FDIM * W2S];    // ~48.75 KB : w2t[f*1040 + e]
  __shared__ float b2s[EDIM];          // 4 KB
  __shared__ float b1s[FPAD];
  __shared__ float ths[FPAD];
  __shared__ float qs[WAVES][256];     // 8 KB : per-wave 16x16 q tile (token-major)

  const int tid   = threadIdx.x;       // 0..255
  const int wave  = tid >> 5;
  const int lane  = tid & 31;
  const int laneM = lane & 15;         // N / M index within half-wave
  const int laneH = lane >> 4;         // half-wave select

  // ---------------- stage weights / biases into LDS (coalesced reads) -------
  for (int i = tid; i < FDIM * EDIM; i += 256) {
    int f = i >> 10, e = i & (EDIM - 1);
    w1t[e * FPAD + f] = W1[i];                     // W1 is (F,E) row-major
  }
  for (int i = tid; i < EDIM * 4; i += 256)
    w1t[(i >> 2) * FPAD + 12 + (i & 3)] = 0.0f;    // zero the f=12..15 pad
  for (int i = tid; i < EDIM * FDIM; i += 256) {
    int e = i / FDIM, f = i - e * FDIM;
    w2t[f * W2S + e] = W2[i];                      // W2 is (E,F) row-major
  }
  for (int i = tid; i < EDIM; i += 256) b2s[i] = b2[i];
  if (tid < FPAD) {
    b1s[tid] = (tid < FDIM) ? b1[tid]    : 0.0f;
    ths[tid] = (tid < FDIM) ? theta[tid] : 0.0f;
  }
  __syncthreads();

  // ---------------- one 16-token tile per wave ------------------------------
  const int  tile    = blockIdx.x * WAVES + wave;  // 0..1023
  const long tokBase = (long)tile * 16;
  // A-operand layout for 16x16x4: lane<16 -> row=lane, K={k0,k0+1};
  //                               lane>=16 -> row=lane-16, K={k0+2,k0+3}
  const float* xrow = x + (tokBase + laneM) * EDIM + 2 * laneH;

  // ---- GEMM1: h(16 tok x 16 f) = X(16x1024) @ W1^T(1024x16), K in chunks of 4
  v8f h0 = {0.f,0.f,0.f,0.f,0.f,0.f,0.f,0.f};
  v8f h1 = h0, h2 = h0, h3 = h0;
  for (int e0 = 0; e0 < EDIM; e0 += 16) {
    {
      v2f a = *(const v2f*)(xrow + e0);
      int kr = e0 + 2 * laneH;
      G_STEP(h0, a, w1t[kr * FPAD + laneM], w1t[(kr + 1) * FPAD + laneM]);
    }
    {
      v2f a = *(const v2f*)(xrow + e0 + 4);
      int kr = e0 + 4 + 2 * laneH;
      G_STEP(h1, a, w1t[kr * FPAD + laneM], w1t[(kr + 1) * FPAD + laneM]);
    }
    {
      v2f a = *(const v2f*)(xrow + e0 + 8);
      int kr = e0 + 8 + 2 * laneH;
      G_STEP(h2, a, w1t[kr * FPAD + laneM], w1t[(kr + 1) * FPAD + laneM]);
    }
    {
      v2f a = *(const v2f*)(xrow + e0 + 12);
      int kr = e0 + 12 + 2 * laneH;
      G_STEP(h3, a, w1t[kr * FPAD + laneM], w1t[(kr + 1) * FPAD + laneM]);
    }
  }
  v8f hacc = (h0 + h1) + (h2 + h3);

  // ---- epilogue: q = cos(relu(h + b1) + theta); lane holds D[tok=v+8*H][f=laneM]
  const float bb1 = b1s[laneM];
  const float tth = ths[laneM];
#pragma unroll
  for (int v = 0; v < 8; ++v) {
    float hv = fmaxf(hacc[v] + bb1, 0.0f);
    float qv = cosf(hv + tth);
    if (laneM >= FDIM) qv = 0.0f;                  // kill padded K columns of GEMM2
    qs[wave][(v + 8 * laneH) * 16 + laneM] = qv;   // token-major 16x16
  }
  asm volatile("s_wait_dscnt 0" ::: "memory");     // wave-local LDS RAW fence

  // Re-load q in A-matrix layout: lane row = laneM (token), K pair = 4c+2*laneH
  v2f aq[3];
#pragma unroll
  for (int c = 0; c < 3; ++c)
    aq[c] = *(const v2f*)&qs[wave][laneM * 16 + 4 * c + 2 * laneH];

  // ---- GEMM2: out(16 tok x 1024) = Q(16x16) @ W2^T(16x1024) + b2, K=12 (3x4)
  for (int e0 = 0; e0 < EDIM; e0 += 16) {
    v8f acc = {0.f,0.f,0.f,0.f,0.f,0.f,0.f,0.f};
#pragma unroll
    for (int c = 0; c < 3; ++c) {
      int kr = 4 * c + 2 * laneH;                  // max 11 < FDIM
      G_STEP(acc, aq[c], w2t[kr * W2S + e0 + laneM], w2t[(kr + 1) * W2S + e0 + laneM]);
    }
    const float bias = b2s[e0 + laneM];
#pragma unroll
    for (int v = 0; v < 8; ++v)
      out[(tokBase + v + 8 * laneH) * EDIM + e0 + laneM] = acc[v] + bias;
  }
}

extern "C" void kernel_launch(void* const* d_in, const int* in_sizes, int n_in,
                              void* d_out, int out_size, void* d_ws, size_t ws_size,
                              hipStream_t stream) {
  const float* x     = (const float*)d_in[0];
  const float* W1    = (const float*)d_in[1];
  const float* b1    = (const float*)d_in[2];
  const float* theta = (const float*)d_in[3];
  const float* W2    = (const float*)d_in[4];
  const float* b2    = (const float*)d_in[5];
  float* out = (float*)d_out;
  (void)in_sizes; (void)n_in; (void)out_size; (void)d_ws; (void)ws_size;

  ffq_fused_wmma<<<dim3(BLOCKS), dim3(256), 0, stream>>>(x, W1, b1, theta, W2, b2, out);
}